// GNNClassifier_37984690766194
// MI455X (gfx1250) — compile-verified
//
#include <hip/hip_runtime.h>
#include <math.h>

#define N_NODES 50000
#define N_EDGES 800000
#define DIM     128
#define NGRAPH  64
#define NCLASS  2
#define ATT_SCALE 0.08838834764831845f  /* 1/sqrt(128) */

typedef float v2f __attribute__((ext_vector_type(2)));
typedef float v8f __attribute__((ext_vector_type(8)));
typedef int   v4i __attribute__((ext_vector_type(4)));
typedef __attribute__((address_space(1))) v4i* gv4i_p;   // global-memory v4i*
typedef __attribute__((address_space(3))) v4i* lv4i_p;   // LDS v4i*

#if defined(__has_builtin)
#if __has_builtin(__builtin_amdgcn_global_load_async_to_lds_b128)
#define HAVE_ASYNC_LDS 1
#endif
#endif

// ---- order-preserving float<->uint keys for atomic max on floats ----
__device__ __forceinline__ unsigned f2ord(float f) {
    unsigned u = __float_as_uint(f);
    return (u & 0x80000000u) ? ~u : (u | 0x80000000u);
}
__device__ __forceinline__ float ord2f(unsigned u) {
    u = (u & 0x80000000u) ? (u & 0x7FFFFFFFu) : ~u;
    return __uint_as_float(u);
}

// ============================================================================
// Fused projection GEMM: O = X @ W^T + b for 4 weight matrices (q,k,v,skip).
// One wave computes a 16(node)x16(feature) tile with V_WMMA_F32_16X16X4_F32,
// looping K=128 in steps of 4. 8 waves/block cover all 8 feature tiles,
// blockIdx.y selects which of the 4 weight matrices. The 16x128 x-tile is
// staged into LDS with CDNA5 async global->LDS copies when available.
// ============================================================================
__global__ __launch_bounds__(256)
void proj_gemm_kernel(const float* __restrict__ X,
                      const float* __restrict__ Wq, const float* __restrict__ Wk,
                      const float* __restrict__ Wv, const float* __restrict__ Ws,
                      const float* __restrict__ bq, const float* __restrict__ bk,
                      const float* __restrict__ bv, const float* __restrict__ bs,
                      float* __restrict__ Oq, float* __restrict__ Ok,
                      float* __restrict__ Ov, float* __restrict__ Os)
{
    __shared__ float xs[16][132];   // padded: conflict-free column reads

    const float* W; const float* B; float* O;
    switch (blockIdx.y) {
        case 0:  W = Wq; B = bq; O = Oq; break;
        case 1:  W = Wk; B = bk; O = Ok; break;
        case 2:  W = Wv; B = bv; O = Ov; break;
        default: W = Ws; B = bs; O = Os; break;
    }

    const int nbase = blockIdx.x * 16;

#if HAVE_ASYNC_LDS
    // 16 rows x 512B: 512 chunks of 16B, 2 per thread, DMA'd straight to LDS.
    {
        #pragma unroll
        for (int it = 0; it < 2; ++it) {
            int chunk = threadIdx.x + it * 256;      // 0..511
            int r  = chunk >> 5;                     // 32 x 16B chunks per row
            int c4 = chunk & 31;                     // 16B unit within row
            const float* g = X + (size_t)(nbase + r) * DIM + c4 * 4;
            __builtin_amdgcn_global_load_async_to_lds_b128(
                (gv4i_p)g, (lv4i_p)&xs[r][c4 * 4], 0, 0);
        }
#if defined(__has_builtin) && __has_builtin(__builtin_amdgcn_s_wait_asynccnt)
        __builtin_amdgcn_s_wait_asynccnt(0);
#else
        asm volatile("s_wait_asynccnt 0x0" ::: "memory");
#endif
    }
#else
    for (int idx = threadIdx.x; idx < 16 * DIM; idx += 256) {
        int r = idx >> 7, c = idx & 127;
        xs[r][c] = X[(size_t)(nbase + r) * DIM + c];
    }
#endif
    __syncthreads();

    const int wave = threadIdx.x >> 5;      // 0..7 -> feature tile
    const int lane = threadIdx.x & 31;
    const int m    = lane & 15;
    const int hv   = lane >> 4;             // half-wave selects K pair
    const int dbase = wave * 16;

    // B[k][n] = W[dbase+n][k0+k]  (since O = X @ W^T)
    const float* wrow = W + (dbase + m) * DIM;

    v8f c = {};
    #pragma unroll
    for (int k0 = 0; k0 < DIM; k0 += 4) {
        v2f a, b;
        a.x = xs[m][k0 + 2 * hv];
        a.y = xs[m][k0 + 2 * hv + 1];
        b.x = wrow[k0 + 2 * hv];
        b.y = wrow[k0 + 2 * hv + 1];
        c = __builtin_amdgcn_wmma_f32_16x16x4_f32(false, a, false, b,
                                                  (short)0, c, false, false);
    }

    const int   col  = dbase + m;
    const float bias = B[col];
    #pragma unroll
    for (int i = 0; i < 8; ++i) {
        int row = nbase + i + (hv ? 8 : 0);   // C/D layout: vgpr i -> M=i / M=i+8
        O[row * DIM + col] = c[i] + bias;
    }
}

// ============================================================================
// Segment-softmax pipeline over edges
// ============================================================================
__global__ void seg_init_kernel(float* __restrict__ AGG, unsigned* __restrict__ amax,
                                float* __restrict__ denom)
{
    int i = blockIdx.x * blockDim.x + threadIdx.x;
    if (i < N_NODES * DIM) AGG[i] = 0.0f;
    if (i < N_NODES) { amax[i] = 0u; denom[i] = 0.0f; }
}

__global__ __launch_bounds__(256)
void edge_alpha_kernel(const int* __restrict__ src, const int* __restrict__ dst,
                       const float* __restrict__ Q, const float* __restrict__ K,
                       float* __restrict__ alpha, unsigned* __restrict__ amax)
{
    const int e = blockIdx.x * 8 + (threadIdx.x >> 5);   // one wave per edge
    if (e >= N_EDGES) return;
    const int lane = threadIdx.x & 31;
    const int s = src[e], d = dst[e];
    const float4 qv = *(const float4*)(Q + (size_t)d * DIM + lane * 4);
    const float4 kv = *(const float4*)(K + (size_t)s * DIM + lane * 4);
    float acc = qv.x * kv.x + qv.y * kv.y + qv.z * kv.z + qv.w * kv.w;
    #pragma unroll
    for (int off = 16; off > 0; off >>= 1) acc += __shfl_xor(acc, off, 32);
    if (lane == 0) {
        float a = acc * ATT_SCALE;
        alpha[e] = a;
        atomicMax(amax + d, f2ord(a));
    }
}

__global__ void edge_exp_kernel(const int* __restrict__ dst, float* __restrict__ alpha,
                                const unsigned* __restrict__ amax, float* __restrict__ denom)
{
    int e = blockIdx.x * blockDim.x + threadIdx.x;
    if (e >= N_EDGES) return;
    int d = dst[e];
    float ea = __expf(alpha[e] - ord2f(amax[d]));
    alpha[e] = ea;                     // reuse buffer for exp(alpha - max)
    atomicAdd(denom + d, ea);
}

__global__ __launch_bounds__(256)
void edge_agg_kernel(const int* __restrict__ src, const int* __restrict__ dst,
                     const float* __restrict__ V, const float* __restrict__ alpha,
                     const float* __restrict__ denom, float* __restrict__ AGG)
{
    const int e = blockIdx.x * 8 + (threadIdx.x >> 5);   // one wave per edge
    if (e >= N_EDGES) return;
    const int lane = threadIdx.x & 31;
    const int s = src[e], d = dst[e];
    const float w = alpha[e] / denom[d];
    const float4 vv = *(const float4*)(V + (size_t)s * DIM + lane * 4);
    float* o = AGG + (size_t)d * DIM + lane * 4;
    atomicAdd(o + 0, vv.x * w);
    atomicAdd(o + 1, vv.y * w);
    atomicAdd(o + 2, vv.z * w);
    atomicAdd(o + 3, vv.w * w);
}

__global__ void skip_relu_kernel(const float* __restrict__ AGG, const float* __restrict__ S,
                                 float* __restrict__ OUT)
{
    int i = blockIdx.x * blockDim.x + threadIdx.x;
    if (i < N_NODES * DIM) OUT[i] = fmaxf(AGG[i] + S[i], 0.0f);
}

// ============================================================================
// Global mean pool + classifier
// ============================================================================
__global__ void pool_zero_kernel(float* __restrict__ psum, float* __restrict__ pcnt)
{
    int i = blockIdx.x * blockDim.x + threadIdx.x;
    if (i < NGRAPH * DIM) psum[i] = 0.0f;
    if (i < NGRAPH) pcnt[i] = 0.0f;
}

__global__ void pool_acc_kernel(const int* __restrict__ batch, const float* __restrict__ H,
                                float* __restrict__ psum, float* __restrict__ pcnt)
{
    int i = blockIdx.x * blockDim.x + threadIdx.x;
    if (i >= N_NODES * DIM) return;
    int n = i >> 7, d = i & 127;
    int g = batch[n];
    atomicAdd(psum + g * DIM + d, H[i]);
    if (d == 0) atomicAdd(pcnt + g, 1.0f);
}

__global__ __launch_bounds__(256)
void classifier_kernel(const float* __restrict__ psum, const float* __restrict__ pcnt,
                       const float* __restrict__ Wc1, const float* __restrict__ bc1,
                       const float* __restrict__ Wc2, const float* __restrict__ bc2,
                       float* __restrict__ out)
{
    __shared__ float pld[NGRAPH * DIM];   // 32 KB
    __shared__ float zld[NGRAPH * 64];    // 16 KB
    for (int i = threadIdx.x; i < NGRAPH * DIM; i += 256) {
        int g = i >> 7;
        pld[i] = psum[i] / fmaxf(pcnt[g], 1.0f);
    }
    __syncthreads();
    for (int i = threadIdx.x; i < NGRAPH * 64; i += 256) {
        int g = i >> 6, j = i & 63;
        float acc = bc1[j];
        for (int k = 0; k < DIM; ++k) acc += pld[g * DIM + k] * Wc1[j * DIM + k];
        zld[i] = fmaxf(acc, 0.0f);
    }
    __syncthreads();
    for (int i = threadIdx.x; i < NGRAPH * NCLASS; i += 256) {
        int g = i >> 1, cc = i & 1;
        float acc = bc2[cc];
        for (int j = 0; j < 64; ++j) acc += zld[g * 64 + j] * Wc2[cc * 64 + j];
        out[i] = acc;
    }
}

// ============================================================================
extern "C" void kernel_launch(void* const* d_in, const int* in_sizes, int n_in,
                              void* d_out, int out_size, void* d_ws, size_t ws_size,
                              hipStream_t stream)
{
    (void)in_sizes; (void)n_in; (void)out_size; (void)ws_size;

    const float* x     = (const float*)d_in[0];
    const int*   ei    = (const int*)d_in[1];
    const int*   src   = ei;
    const int*   dst   = ei + N_EDGES;
    const int*   batch = (const int*)d_in[2];

    const float* Wl[2][4];
    const float* bl[2][4];
    for (int l = 0; l < 2; ++l) {
        for (int i = 0; i < 4; ++i) Wl[l][i] = (const float*)d_in[3 + l * 8 + i];
        for (int i = 0; i < 4; ++i) bl[l][i] = (const float*)d_in[3 + l * 8 + 4 + i];
    }
    const float* Wc1 = (const float*)d_in[19];
    const float* bc1 = (const float*)d_in[20];
    const float* Wc2 = (const float*)d_in[21];
    const float* bc2 = (const float*)d_in[22];
    float* out = (float*)d_out;

    // ---- workspace carve-out (floats) ----
    const size_t nd = (size_t)N_NODES * DIM;
    float* ws = (float*)d_ws;
    float* Q     = ws; ws += nd;
    float* K     = ws; ws += nd;
    float* V     = ws; ws += nd;
    float* S     = ws; ws += nd;
    float* AGG   = ws; ws += nd;
    float* H     = ws; ws += nd;
    float* ALPHA = ws; ws += N_EDGES;
    unsigned* AMAX = (unsigned*)ws; ws += N_NODES;
    float* DENOM = ws; ws += N_NODES;
    float* PSUM  = ws; ws += NGRAPH * DIM;
    float* PCNT  = ws; ws += NGRAPH;

    const dim3 gemm_grid(N_NODES / 16, 4);           // 3125 x 4
    const int nd_blocks        = (int)((nd + 255) / 256);
    const int edge_wave_blocks = (N_EDGES + 7) / 8;  // one wave per edge
    const int edge_blocks      = (N_EDGES + 255) / 256;

    for (int l = 0; l < 2; ++l) {
        const float* in = (l == 0) ? x : H;
        proj_gemm_kernel<<<gemm_grid, 256, 0, stream>>>(in,
            Wl[l][0], Wl[l][1], Wl[l][2], Wl[l][3],
            bl[l][0], bl[l][1], bl[l][2], bl[l][3],
            Q, K, V, S);
        seg_init_kernel<<<nd_blocks, 256, 0, stream>>>(AGG, AMAX, DENOM);
        edge_alpha_kernel<<<edge_wave_blocks, 256, 0, stream>>>(src, dst, Q, K, ALPHA, AMAX);
        edge_exp_kernel<<<edge_blocks, 256, 0, stream>>>(dst, ALPHA, AMAX, DENOM);
        edge_agg_kernel<<<edge_wave_blocks, 256, 0, stream>>>(src, dst, V, ALPHA, DENOM, AGG);
        // layer 0 result -> H (input of layer 1); layer 1 result -> AGG in place
        skip_relu_kernel<<<nd_blocks, 256, 0, stream>>>(AGG, S, (l == 0) ? H : AGG);
    }

    pool_zero_kernel<<<(NGRAPH * DIM + 255) / 256, 256, 0, stream>>>(PSUM, PCNT);
    pool_acc_kernel<<<nd_blocks, 256, 0, stream>>>(batch, AGG, PSUM, PCNT);
    classifier_kernel<<<1, 256, 0, stream>>>(PSUM, PCNT, Wc1, bc1, Wc2, bc2, out);
}